// End2EndTagLoss_87136296501780
// MI455X (gfx1250) — compile-verified
//
#include <hip/hip_runtime.h>
#include <stdint.h>

#define IMGS     4
#define NPROP    2000
#define NPAD     2048
#define NTHREADS 512
#define NWAVES   (NTHREADS / 32)
#define ITEMS    (NPAD / NTHREADS)   // 4 elements per thread, stride-NTHREADS interleave
#define TAG_THR  0.1f
#define LOSS_EPS 1e-6f

typedef float v2f __attribute__((ext_vector_type(2)));
typedef float v8f __attribute__((ext_vector_type(8)));
typedef float f4  __attribute__((ext_vector_type(4)));

// ---------------- LDS layout (bytes) ----------------
#define OFF_PRED 0                       // f4[NPAD]   32768 (16B aligned)
#define OFF_BOX  (OFF_PRED + NPAD * 16)  // f4[NPAD]   32768 (x1,y1,x2,y2)
#define OFF_SC   (OFF_BOX + NPAD * 16)   // f32[NPAD]
#define OFF_NRM  (OFF_SC + NPAD * 4)     // f32[NPAD]  |p_j|^2 (WMMA Gram diag)
#define OFF_GT   (OFF_NRM + NPAD * 4)    // i32[NPAD]
#define OFF_SRK  (OFF_GT + NPAD * 4)     // i32[NPAD]  rank while alive, -1 dead
#define OFF_ORD  (OFF_SRK + NPAD * 4)    // i32[NPAD]  sorted order (desc score)
#define OFF_KEY  (OFF_ORD + NPAD * 4)    // u64[NPAD]  sort keys
#define OFF_RED  (OFF_KEY + NPAD * 8)    // 2 x (2f + 3i) x 16 slots, dbl-buffered
#define SMEM_BYTES (OFF_RED + 5 * 2 * 16 * 4)   // ~120.6 KB << 320 KB WGP LDS

__device__ __forceinline__ uint32_t lds_of(const void* p) {
  // generic pointer derived from __shared__: low 32 bits are the LDS byte offset
  return (uint32_t)(uintptr_t)p;
}

// CDNA5 async global->LDS (ASYNCcnt-tracked), GVS addressing: saddr(64) + vaddr(32)
__device__ __forceinline__ void async_g2l_b32(uint32_t lds_addr, uint32_t goff, const void* base) {
  asm volatile("global_load_async_to_lds_b32 %0, %1, %2"
               :: "v"(lds_addr), "v"(goff), "s"(base) : "memory");
}
__device__ __forceinline__ void async_g2l_b128(uint32_t lds_addr, uint32_t goff, const void* base) {
  asm volatile("global_load_async_to_lds_b128 %0, %1, %2"
               :: "v"(lds_addr), "v"(goff), "s"(base) : "memory");
}
__device__ __forceinline__ void wait_async0() {
  asm volatile("s_wait_asynccnt 0" ::: "memory");
}

__global__ __launch_bounds__(NTHREADS)
void tag_loss_kernel(const float* __restrict__ pred,
                     const int* __restrict__ gt_inds,
                     const float* __restrict__ prop,
                     float* __restrict__ ws /* IMGS*2: push,pull per image */) {
  extern __shared__ char smem[];
  f4*       spred = (f4*)(smem + OFF_PRED);
  f4*       sbox  = (f4*)(smem + OFF_BOX);
  float*    sboxf = (float*)(smem + OFF_BOX);
  float*    ssc   = (float*)(smem + OFF_SC);
  float*    snorm = (float*)(smem + OFF_NRM);
  int*      sgt   = (int*)(smem + OFF_GT);
  int*      srk   = (int*)(smem + OFF_SRK);
  int*      sord  = (int*)(smem + OFF_ORD);
  uint64_t* skey  = (uint64_t*)(smem + OFF_KEY);
  float*    pf0   = (float*)(smem + OFF_RED);        // pull partials [2][16]
  float*    pf1   = pf0 + 32;                        // push partials
  int*      pi0   = (int*)(pf1 + 32);                // pn
  int*      pi1   = pi0 + 32;                        // qn
  int*      pi2   = pi1 + 32;                        // #suppressed

  const int img = blockIdx.x;
  const int tid = threadIdx.x;
  const int wv  = tid >> 5;
  const int ln  = tid & 31;

  const float* predI = pred + (size_t)img * NPROP * 4;
  const int*   gtI   = gt_inds + (size_t)img * NPROP;
  const float* prI   = prop + (size_t)img * NPROP * 5;

  // ---- Stage 0: async global->LDS fill with free AoS->SoA deinterleave ----
  for (int j = tid; j < NPROP; j += NTHREADS) {
    async_g2l_b128(lds_of(&spred[j]), (uint32_t)(j * 16), predI);
    async_g2l_b32(lds_of(&sboxf[j * 4 + 0]), (uint32_t)((j * 5 + 0) * 4), prI);
    async_g2l_b32(lds_of(&sboxf[j * 4 + 1]), (uint32_t)((j * 5 + 1) * 4), prI);
    async_g2l_b32(lds_of(&sboxf[j * 4 + 2]), (uint32_t)((j * 5 + 2) * 4), prI);
    async_g2l_b32(lds_of(&sboxf[j * 4 + 3]), (uint32_t)((j * 5 + 3) * 4), prI);
    async_g2l_b32(lds_of(&ssc[j]), (uint32_t)((j * 5 + 4) * 4), prI);
    async_g2l_b32(lds_of(&sgt[j]), (uint32_t)(j * 4), gtI);
  }
  wait_async0();
  __syncthreads();

  // ---- Stage 1: sort keys (score desc, index asc) + WMMA Gram-diag norms ----
  for (int j = tid; j < NPAD; j += NTHREADS) {
    const float s = (j < NPROP) ? ssc[j] : 0.0f;       // scores are >= 0
    skey[j] = ((uint64_t)__float_as_uint(s) << 32) | (uint32_t)(~j);
  }

#if __has_builtin(__builtin_amdgcn_wmma_f32_16x16x4_f32)
  // A = B = pred tile: for f32 16x4 A-layout and 4x16 B-layout the per-lane
  // register contents coincide. D[m][m] = <p_{base+m}, p_{base+m}>.
  for (int T = wv; T < NPAD / 16; T += NWAVES) {
    const int base = T * 16;
    f4 p4 = spred[base + (ln & 15)];
    v2f a;
    a.x = (ln < 16) ? p4.x : p4.z;   // K0 / K2
    a.y = (ln < 16) ? p4.y : p4.w;   // K1 / K3
    v8f c = {0.f, 0.f, 0.f, 0.f, 0.f, 0.f, 0.f, 0.f};
    v8f d = __builtin_amdgcn_wmma_f32_16x16x4_f32(false, a, false, a,
                                                  (short)0, c, false, false);
    int m = -1, v = 0;
    if (ln < 8)        { m = ln;      v = ln;      }
    else if (ln >= 24) { m = ln - 16; v = ln - 24; }
    if (m >= 0) {
      float val = 0.f;
#pragma unroll
      for (int k = 0; k < 8; ++k) if (k == v) val = d[k];
      snorm[base + m] = val;
    }
  }
#else
  for (int j = tid; j < NPROP; j += NTHREADS) {
    f4 pp = spred[j];
    snorm[j] = pp.x * pp.x + pp.y * pp.y + pp.z * pp.z + pp.w * pp.w;
  }
#endif
  __syncthreads();

  // ---- Stage 2: bitonic sort, descending by key (one-time cost) ----
  for (int k = 2; k <= NPAD; k <<= 1) {
    for (int l = k >> 1; l > 0; l >>= 1) {
      for (int t = tid; t < NPAD / 2; t += NTHREADS) {
        const int a = ((t & ~(l - 1)) << 1) | (t & (l - 1));
        const int b = a | l;
        const uint64_t ka = skey[a], kb = skey[b];
        if ((ka < kb) == ((a & k) == 0)) { skey[a] = kb; skey[b] = ka; }
      }
      __syncthreads();
    }
  }

  // order + rank-encoded alive state
  for (int q = tid; q < NPAD; q += NTHREADS) {
    const int j = (int)(~(uint32_t)skey[q]);
    sord[q] = j;
    const bool valid = (j < NPROP) && (sgt[j] >= 0) &&
                       ((sboxf[j * 4 + 3] - sboxf[j * 4 + 1]) > 0.0f);
    srk[j] = valid ? q : -1;
  }
  __syncthreads();

  // ---- Stage 3: initial alive count (uses parity-0 slots) ----
  {
    int c = 0;
#pragma unroll
    for (int kk = 0; kk < ITEMS; ++kk) c += (srk[tid + kk * NTHREADS] >= 0);
#pragma unroll
    for (int off = 16; off; off >>= 1) c += __shfl_xor(c, off);
    if (ln == 0) pi2[wv] = c;   // parity-0 region of pi2
  }
  __syncthreads();
  int cnt;
  {
    int c = (ln < NWAVES) ? pi2[ln] : 0;
#pragma unroll
    for (int off = 16; off; off >>= 1) c += __shfl_xor(c, off);
    cnt = c;
  }

  // registers, identical in every thread (deterministic)
  float tot_pull = 0.f, tot_push = 0.f, pull_cnt = 0.f, push_cnt = 0.f;
  int p = -1, i = 0, par = 1;
  if (cnt >= 2) {
    int q = 0;
    while (q < NPAD && srk[sord[q]] != q) ++q;   // broadcast LDS reads
    p = q;
    i = (q < NPAD) ? sord[q] : 0;
  }

  // ---- Stage 4: greedy scan — ONE barrier per iteration ----
  for (int iter = 0; iter < NPROP; ++iter) {
    if (cnt < 2) break;   // reference state freezes from here on

    const f4    bi = sbox[i];
    const f4    pi = spred[i];
    const float ni = snorm[i];
    const int   gi = sgt[i];

    float pull = 0.f, push = 0.f;
    int pn = 0, qn = 0, ncl = 0;
#pragma unroll
    for (int kk = 0; kk < ITEMS; ++kk) {
      const int j = tid + kk * NTHREADS;
      if (srk[j] <= p) continue;                 // dead/suppressed/former-i/i
      const f4 bj = sbox[j];
      const float iw = fminf(bi.z, bj.z) - fmaxf(bi.x, bj.x);
      const float ih = fminf(bi.w, bj.w) - fmaxf(bi.y, bj.y);
      if (!(iw > 0.f && ih > 0.f)) continue;     // iou > 0 <=> iw>0 && ih>0
      const f4 pj = spred[j];
      const float dot  = pi.x * pj.x + pi.y * pj.y + pi.z * pj.z + pi.w * pj.w;
      const float diff = fmaxf(0.25f * (ni + snorm[j]) - 0.5f * dot, 0.f);
      const bool check = diff < TAG_THR;
      const bool same  = (sgt[j] == gi);
      const float sc = ssc[j];
      if (same && !check) { pull += diff * sc; ++pn; }
      if (!same && check) { push += expf(-diff) * sc; ++qn; }
      if (check) { srk[j] = -1; ++ncl; }         // only ranks > p get cleared
    }

    // wave32 reduce of 5 values
#pragma unroll
    for (int off = 16; off; off >>= 1) {
      pull += __shfl_xor(pull, off);
      push += __shfl_xor(push, off);
      pn   += __shfl_xor(pn, off);
      qn   += __shfl_xor(qn, off);
      ncl  += __shfl_xor(ncl, off);
    }
    const int sb = par * 16 + wv;
    if (ln == 0) { pf0[sb] = pull; pf1[sb] = push; pi0[sb] = pn; pi1[sb] = qn; pi2[sb] = ncl; }
    __syncthreads();                              // the ONLY barrier per step

    // every thread re-reduces the 16 per-wave slots (double-buffered by parity)
    {
      const int rb = par * 16 + ln;
      float P = (ln < NWAVES) ? pf0[rb] : 0.f;
      float Q = (ln < NWAVES) ? pf1[rb] : 0.f;
      int   a = (ln < NWAVES) ? pi0[rb] : 0;
      int   b = (ln < NWAVES) ? pi1[rb] : 0;
      int   n = (ln < NWAVES) ? pi2[rb] : 0;
#pragma unroll
      for (int off = 16; off; off >>= 1) {
        P += __shfl_xor(P, off);
        Q += __shfl_xor(Q, off);
        a += __shfl_xor(a, off);
        b += __shfl_xor(b, off);
        n += __shfl_xor(n, off);
      }
      if (a > 0) { tot_pull += P / (float)a; pull_cnt += 1.f; }
      if (b > 0) { tot_push += Q / (float)b; push_cnt += 1.f; }
      cnt -= (1 + n);
    }

    // advance pointer: first alive in sorted order (amortized O(1))
    if (cnt >= 2) {
      int q = p + 1;
      while (q < NPAD && srk[sord[q]] != q) ++q;
      p = q;
      i = (q < NPAD) ? sord[q] : 0;
    }
    par ^= 1;
  }

  if (tid == 0) {
    ws[img * 2 + 0] = tot_push / (push_cnt + LOSS_EPS);  // push
    ws[img * 2 + 1] = tot_pull / (pull_cnt + LOSS_EPS);  // pull
  }
}

__global__ void finalize_kernel(const float* __restrict__ ws, float* __restrict__ out) {
  if (threadIdx.x == 0 && blockIdx.x == 0) {
    float ps = 0.f, pl = 0.f;
    for (int k = 0; k < IMGS; ++k) { ps += ws[2 * k + 0]; pl += ws[2 * k + 1]; }
    out[0] = ps * (1.0f / IMGS);  // push_loss (PUSH_WEIGHT = 1)
    out[1] = pl * (1.0f / IMGS);  // pull_loss (PULL_WEIGHT = 1)
  }
}

extern "C" void kernel_launch(void* const* d_in, const int* in_sizes, int n_in,
                              void* d_out, int out_size, void* d_ws, size_t ws_size,
                              hipStream_t stream) {
  const float* pred = (const float*)d_in[0];   // [4,2000,4] f32
  const int*   gt   = (const int*)d_in[1];     // [4,2000]   i32
  const float* prop = (const float*)d_in[2];   // [4,2000,5] f32
  // d_in[3] (gt_bboxes) is unused by the reference loss.
  float* ws  = (float*)d_ws;
  float* out = (float*)d_out;

  tag_loss_kernel<<<dim3(IMGS), dim3(NTHREADS), SMEM_BYTES, stream>>>(pred, gt, prop, ws);
  finalize_kernel<<<dim3(1), dim3(32), 0, stream>>>(ws, out);
}